// RecurrentDNNCNoFalsePop_86474871537849
// MI455X (gfx1250) — compile-verified
//
#include <hip/hip_runtime.h>

// ---------------------------------------------------------------------------
// RecurrentDNN as a (max,+) parallel prefix scan.
//   h_t = w1.x_t + b1                        (via V_WMMA_F32_16X16X4_F32)
//   y_t = max(y_{t-1} + h_t, 0)              (op (a,b): y -> max(y+a, b))
//   out = sigmoid(w2*y + b2)
// compose((a1,b1),(a2,b2)) = (a1+a2, max(b1+a2, b2)); identity (0, -inf)
// ---------------------------------------------------------------------------

typedef __attribute__((ext_vector_type(2))) float v2f;
typedef __attribute__((ext_vector_type(8))) float v8f;

#define WAVE        32
#define THREADS_P13 256
#define BLOCKS_P13  1024
#define WAVES_TOTAL (BLOCKS_P13 * (THREADS_P13 / WAVE))   // 8192
#define THREADS_P2  1024
#define ITEMS_P2    (WAVES_TOTAL / THREADS_P2)            // 8

static __device__ __forceinline__ float neg_inf() { return -__builtin_inff(); }

struct Op {
  float a, b;
};
// compose(first-in-time p, later q)
static __device__ __forceinline__ Op comp(Op p, Op q) {
  Op r;
  r.a = p.a + q.a;
  r.b = fmaxf(p.b + q.a, q.b);
  return r;
}
static __device__ __forceinline__ Op op_ident() {
  Op r; r.a = 0.0f; r.b = neg_inf(); return r;
}
static __device__ __forceinline__ Op shfl_xor_op(Op v, int mask) {
  Op r;
  r.a = __shfl_xor(v.a, mask, WAVE);
  r.b = __shfl_xor(v.b, mask, WAVE);
  return r;
}
// Branchless ordered compose: if x_first -> x∘y else y∘x.
// 'a' is commutative; compute both 'b' orders and v_cndmask one.
static __device__ __forceinline__ Op ordered_comp(Op x, Op y, bool x_first) {
  Op r;
  r.a = x.a + y.a;
  const float b_xy = fmaxf(x.b + y.a, y.b);
  const float b_yx = fmaxf(y.b + x.a, x.b);
  r.b = x_first ? b_xy : b_yx;
  return r;
}

// D = A(16x4) * B(4x16) + C, f32. A rows m carry TWO timesteps:
//   A[m][0..1] = x pair of (t0+m),  A[m][2..3] = x pair of (t0+16+m)
// B columns 0..7 select the K=0,1 pair, columns 8..15 select K=2,3.
// Lane n then holds h_{t0 + 8*s + r} in d[r], s = time-segment of the lane.
static __device__ __forceinline__ v8f wmma_h(v2f a, v2f b, float bias) {
  v8f c = {bias, bias, bias, bias, bias, bias, bias, bias};
#if defined(__HIP_DEVICE_COMPILE__)
#if __has_builtin(__builtin_amdgcn_wmma_f32_16x16x4_f32)
  c = __builtin_amdgcn_wmma_f32_16x16x4_f32(false, a, false, b, (short)0, c,
                                            false, false);
#else
  asm volatile("v_wmma_f32_16x16x4_f32 %0, %1, %2, %0"
               : "+v"(c)
               : "v"(a), "v"(b));
#endif
#endif
  return c;
}

// depth-3 tree composition of the lane's contiguous 8-run (aggregate only)
static __device__ __forceinline__ Op run8_agg(v8f d) {
  Op e0, e1, e2, e3, e4, e5, e6, e7;
#define MKOP(E, H) { E.a = (H); E.b = fmaxf((H), 0.0f); }
  MKOP(e0, d[0]) MKOP(e1, d[1]) MKOP(e2, d[2]) MKOP(e3, d[3])
  MKOP(e4, d[4]) MKOP(e5, d[5]) MKOP(e6, d[6]) MKOP(e7, d[7])
#undef MKOP
  Op l0 = comp(e0, e1), l1 = comp(e2, e3), l2 = comp(e4, e5), l3 = comp(e6, e7);
  return comp(comp(l0, l1), comp(l2, l3));
}

// Ordered xor-butterfly across the 4 time-segments (fully branchless).
//   lanes [0-7]=s0, [16-23]=s1, [8-15]=s2, [24-31]=s3
//   step1: xor 16 pairs (s0,s1) and (s2,s3)  (adjacent in time)
//   step2: xor  8 pairs (s01,s23)
// Outputs: total of all 32 elements + exclusive prefix for this lane's
// segment.  excl(s): s0 -> I, s1 -> o1, s2 -> o2, s3 -> o2∘o1, expressed as
// base = (hiN ? o2 : I) then post-compose with o1 iff hiK.
static __device__ __forceinline__ void seg_combine(Op self, bool hiK, bool hiN,
                                                   Op* total, Op* excl) {
  Op o1 = shfl_xor_op(self, 16);            // partner single-segment agg
  Op p  = ordered_comp(self, o1, !hiK);     // time-pair aggregate
  Op o2 = shfl_xor_op(p, 8);                // other pair aggregate
  *total = ordered_comp(p, o2, !hiN);

  float ea = hiN ? o2.a : 0.0f;
  float eb = hiN ? o2.b : neg_inf();
  const float ea2 = ea + o1.a;
  const float eb2 = fmaxf(eb + o1.a, o1.b);
  excl->a = hiK ? ea2 : ea;
  excl->b = hiK ? eb2 : eb;
}

// ---------------------------------------------------------------------------
// Phase 1: per-wave operator aggregates (register double-buffered stream).
// ---------------------------------------------------------------------------
__global__ void __launch_bounds__(THREADS_P13)
rdnn_phase1(const float* __restrict__ x, const float* __restrict__ w1,
            const float* __restrict__ b1, float* __restrict__ segA,
            float* __restrict__ segB, int segLen) {
  const int lane = threadIdx.x & (WAVE - 1);
  const int wave = (int)((blockIdx.x * blockDim.x + threadIdx.x) >> 5);
  const long t0base = (long)wave * (long)segLen;

  const float w10 = w1[0], w11 = w1[1], bb1 = b1[0];
  const bool hiK = lane >= 16;
  const bool hiN = (lane & 15) >= 8;
  v2f Bv;
  Bv.x = (hiK == hiN) ? w10 : 0.0f;
  Bv.y = (hiK == hiN) ? w11 : 0.0f;

  const float2* __restrict__ xp = (const float2*)x;
  Op agg = op_ident();

  float2 xv = xp[t0base + lane];  // prime the pipeline
  for (int it = 0; it < segLen; it += 32) {
    float2 cur = xv;
    const int itn = (it + 32 < segLen) ? (it + 32) : it;  // branchless clamp
    xv = xp[t0base + itn + lane];                         // next tile in flight
    __builtin_prefetch(&xp[t0base + it + lane + 256], 0, 1);

    v2f Av; Av.x = cur.x; Av.y = cur.y;
    v8f d = wmma_h(Av, Bv, bb1);

    Op self = run8_agg(d);
    Op total, excl;
    seg_combine(self, hiK, hiN, &total, &excl);
    agg = comp(agg, total);
  }
  if (lane == 0) {
    segA[wave] = agg.a;
    segB[wave] = agg.b;
  }
}

// ---------------------------------------------------------------------------
// Phase 2: exclusive scan of the 8192 wave aggregates -> per-wave carry y.
// ---------------------------------------------------------------------------
__global__ void __launch_bounds__(THREADS_P2)
rdnn_phase2(const float* __restrict__ segA, const float* __restrict__ segB,
            float* __restrict__ carry) {
  __shared__ float sa[THREADS_P2];
  __shared__ float sb[THREADS_P2];
  const int t = threadIdx.x;
  const int base = t * ITEMS_P2;

  float la[ITEMS_P2], lb[ITEMS_P2];
  Op th = op_ident();
#pragma unroll
  for (int j = 0; j < ITEMS_P2; ++j) {
    la[j] = segA[base + j];
    lb[j] = segB[base + j];
    Op e; e.a = la[j]; e.b = lb[j];
    th = comp(th, e);
  }
  sa[t] = th.a;
  sb[t] = th.b;
  __syncthreads();
  for (int off = 1; off < THREADS_P2; off <<= 1) {
    Op p = op_ident();
    const bool has = (t >= off);
    if (has) { p.a = sa[t - off]; p.b = sb[t - off]; }
    __syncthreads();
    if (has) {
      th = comp(p, th);
      sa[t] = th.a;
      sb[t] = th.b;
    }
    __syncthreads();
  }
  Op e = op_ident();
  if (t > 0) { e.a = sa[t - 1]; e.b = sb[t - 1]; }
#pragma unroll
  for (int j = 0; j < ITEMS_P2; ++j) {
    carry[base + j] = fmaxf(e.a, e.b);  // f_excl(0)
    Op s; s.a = la[j]; s.b = lb[j];
    e = comp(e, s);
  }
}

// ---------------------------------------------------------------------------
// Phase 3: re-stream x, apply carries, emit sigmoid(w2*y + b2).
// ---------------------------------------------------------------------------
__global__ void __launch_bounds__(THREADS_P13)
rdnn_phase3(const float* __restrict__ x, const float* __restrict__ w1,
            const float* __restrict__ b1, const float* __restrict__ w2,
            const float* __restrict__ b2, const float* __restrict__ carry,
            float* __restrict__ out, int segLen) {
  const int lane = threadIdx.x & (WAVE - 1);
  const int wave = (int)((blockIdx.x * blockDim.x + threadIdx.x) >> 5);
  const long t0base = (long)wave * (long)segLen;

  const float w10 = w1[0], w11 = w1[1], bb1 = b1[0];
  const float w20 = w2[0], bb2 = b2[0];
  const bool hiK = lane >= 16;
  const bool hiN = (lane & 15) >= 8;
  v2f Bv;
  Bv.x = (hiK == hiN) ? w10 : 0.0f;
  Bv.y = (hiK == hiN) ? w11 : 0.0f;
  const int s = (hiK ? 1 : 0) + (hiN ? 2 : 0);  // time-order segment index
  const int myIdx = lane & 7;

  const float2* __restrict__ xp = (const float2*)x;
  float y = carry[wave];

  float2 xv = xp[t0base + lane];
  for (int it = 0; it < segLen; it += 32) {
    float2 cur = xv;
    const int itn = (it + 32 < segLen) ? (it + 32) : it;
    xv = xp[t0base + itn + lane];
    __builtin_prefetch(&xp[t0base + it + lane + 256], 0, 1);

    v2f Av; Av.x = cur.x; Av.y = cur.y;
    v8f d = wmma_h(Av, Bv, bb1);

    Op self = run8_agg(d);
    Op total, excl;
    seg_combine(self, hiK, hiN, &total, &excl);

    float ys = fmaxf(y + excl.a, excl.b);  // value entering this 8-run
    float yk = 0.0f;
#pragma unroll
    for (int r = 0; r < 8; ++r) {
      ys = fmaxf(ys + d[r], 0.0f);
      yk = (r == myIdx) ? ys : yk;
    }
    float z = fmaf(w20, yk, bb2);
    out[t0base + it + 8 * s + myIdx] = 1.0f / (1.0f + __expf(-z));

    y = fmaxf(y + total.a, total.b);  // advance wave carry
  }
}

// ---------------------------------------------------------------------------
extern "C" void kernel_launch(void* const* d_in, const int* in_sizes, int n_in,
                              void* d_out, int out_size, void* d_ws,
                              size_t ws_size, hipStream_t stream) {
  (void)n_in; (void)out_size; (void)ws_size;
  const float* x  = (const float*)d_in[0];
  const float* w1 = (const float*)d_in[1];
  const float* b1 = (const float*)d_in[2];
  const float* w2 = (const float*)d_in[3];
  const float* b2 = (const float*)d_in[4];
  float* out = (float*)d_out;

  const int N = in_sizes[0] / 2;       // SEQ_LEN (x is [1,N,2])
  const int segLen = N / WAVES_TOTAL;  // 1024 for N = 2^23

  float* segA  = (float*)d_ws;         // 8192 floats
  float* segB  = segA + WAVES_TOTAL;   // 8192 floats
  float* carry = segB + WAVES_TOTAL;   // 8192 floats (96 KB total)

  rdnn_phase1<<<BLOCKS_P13, THREADS_P13, 0, stream>>>(x, w1, b1, segA, segB,
                                                      segLen);
  rdnn_phase2<<<1, THREADS_P2, 0, stream>>>(segA, segB, carry);
  rdnn_phase3<<<BLOCKS_P13, THREADS_P13, 0, stream>>>(x, w1, b1, w2, b2, carry,
                                                      out, segLen);
}